// SplitPool_41824391528701
// MI455X (gfx1250) — compile-verified
//
#include <hip/hip_runtime.h>

typedef float v2f __attribute__((ext_vector_type(2)));
typedef float v8f __attribute__((ext_vector_type(8)));

#define DIM       128
#define TILE_ROWS 64
#define LDSP      136   // padded LDS row stride (floats): rows r, r+2 hit disjoint banks
#define NTHREADS  256

#if __has_builtin(__builtin_amdgcn_global_load_async_to_lds_b128) && \
    __has_builtin(__builtin_amdgcn_s_wait_asynccnt)
#define ASYNC_OK 1
#else
#define ASYNC_OK 0
#endif

#if ASYNC_OK
typedef int v4i __attribute__((vector_size(16)));
typedef __attribute__((address_space(1))) v4i* as1_v4i;   // global src
typedef __attribute__((address_space(3))) v4i* as3_v4i;   // LDS dst
#endif

// ---------------- prep: prefix scans (row offsets per chunk, pool starts) ----
__global__ void SplitPool_prep_kernel(const int* __restrict__ chunk_size,
                                      const int* __restrict__ n_peaks,
                                      int* __restrict__ row_start,   // [total_chunks+1]
                                      int* __restrict__ pool_start,  // [batch+1]
                                      int total_chunks, int batch) {
  __shared__ int partial[NTHREADS];
  const int t   = threadIdx.x;
  const int per = (total_chunks + NTHREADS - 1) / NTHREADS;
  const int beg = t * per;
  const int end = min(beg + per, total_chunks);

  int s = 0;
  for (int i = beg; i < end; ++i) s += chunk_size[i];
  partial[t] = s;
  __syncthreads();
  if (t == 0) {
    int acc = 0;
    for (int i = 0; i < NTHREADS; ++i) { int v = partial[i]; partial[i] = acc; acc += v; }
  }
  __syncthreads();
  int acc = partial[t];
  for (int i = beg; i < end; ++i) { row_start[i] = acc; acc += chunk_size[i]; }
  if (beg < total_chunks && end == total_chunks) row_start[total_chunks] = acc;

  if (t == 0) {
    int a = 0; pool_start[0] = 0;
    for (int b = 0; b < batch; ++b) { a += n_peaks[b] + 1; pool_start[b + 1] = a; }
  }
}

// ---------------- zero the per-chunk fp32 accumulators -----------------------
__global__ void SplitPool_zero_kernel(float* __restrict__ p, long n) {
  long i = (long)blockIdx.x * blockDim.x + threadIdx.x;
  if (i < n) p[i] = 0.0f;
}

// ---------------- main: LDS-staged segment-sum via V_WMMA_F32_16X16X4_F32 ----
__global__ void __launch_bounds__(NTHREADS)
SplitPool_segsum_wmma_kernel(const float* __restrict__ x,
                             const int* __restrict__ row_start,
                             float* __restrict__ sums,
                             int total_chunks) {
  __shared__ float tile[TILE_ROWS * LDSP];
  __shared__ int   segs[TILE_ROWS];

  const int  t       = threadIdx.x;
  const long rowBase = (long)blockIdx.x * TILE_ROWS;
  const float* tileSrc = x + rowBase * DIM;   // 64 consecutive rows: contiguous 32KB

  // ---- stage tile into LDS: 16B per lane, fully coalesced -------------------
  // float4 chunk ci: row = ci>>5, cols 4*(ci&31)..+3
  #pragma unroll
  for (int j = 0; j < (TILE_ROWS * DIM / 4) / NTHREADS; ++j) {
    const int ci  = t + j * NTHREADS;
    const int row = ci >> 5;
    const int c4  = (ci & 31) << 2;
#if ASYNC_OK
    __builtin_amdgcn_global_load_async_to_lds_b128(
        (as1_v4i)(tileSrc + ci * 4), (as3_v4i)(&tile[row * LDSP + c4]), 0, 0);
#else
    const float4 v = *(const float4*)(tileSrc + ci * 4);
    *(float4*)(&tile[row * LDSP + c4]) = v;
#endif
  }

  // ---- overlap: per-row segment id via binary search on row_start -----------
  if (t < TILE_ROWS) {
    const long r = rowBase + t;
    int lo = 0, hi = total_chunks - 1;
    while (lo < hi) {
      int mid = (lo + hi + 1) >> 1;
      if ((long)row_start[mid] <= r) lo = mid; else hi = mid - 1;
    }
    segs[t] = lo;
  }

#if ASYNC_OK
  __builtin_amdgcn_s_wait_asynccnt(0);
#endif
  __syncthreads();

  const int  lane    = t & 31;
  const int  wave    = t >> 5;          // 8 waves: one 16-column tile each
  const int  colBase = wave * 16;
  const int  nLo     = lane & 15;       // N (cols) and M (segment slot) index
  const int  hiHalf  = (lane >= 16);
  const int  kOff    = hiHalf ? 2 : 0;  // K rows held by this lane half

  v8f c;
  #pragma unroll
  for (int v = 0; v < 8; ++v) c[v] = 0.0f;

  int segLo = segs[0];   // base of current 16-segment accumulation window
  int segHi = segLo;     // highest segment touched so far in the window

  const float* tcol = &tile[colBase + nLo];

  for (int r0 = 0; r0 < TILE_ROWS; r0 += 4) {
    const int sFirst = segs[r0];
    const int sLast  = segs[r0 + 3];          // segs monotonic within tile
    const int sA     = segs[r0 + kOff];       // segment of row for a.x
    const int sB     = segs[r0 + kOff + 1];   // segment of row for a.y

    if (sLast - segLo >= 16) {                // window overflow: flush + slide
      #pragma unroll
      for (int v = 0; v < 8; ++v) {
        const int seg = segLo + v + (hiHalf ? 8 : 0);
        if (seg <= segHi) atomicAdd(&sums[(long)seg * DIM + colBase + nLo], c[v]);
        c[v] = 0.0f;
      }
      segLo = sFirst;
    }
    segHi = sLast;

    // A (16x4 mask): A[m,k] = 1 iff row (r0+k) belongs to local segment m
    v2f a;
    a.x = ((sA - segLo) == nLo) ? 1.0f : 0.0f;
    a.y = ((sB - segLo) == nLo) ? 1.0f : 0.0f;

    // B (4x16) from LDS: rows r0+kOff, r0+kOff+1 of this wave's column tile
    const float* p = tcol + (r0 + kOff) * LDSP;
    v2f b;
    b.x = p[0];
    b.y = p[LDSP];

    // EXEC is all-ones here (uniform control flow) as the ISA requires.
    c = __builtin_amdgcn_wmma_f32_16x16x4_f32(false, a, false, b,
                                              (short)0, c, false, false);
  }

  #pragma unroll
  for (int v = 0; v < 8; ++v) {
    const int seg = segLo + v + (hiHalf ? 8 : 0);
    if (seg <= segHi) atomicAdd(&sums[(long)seg * DIM + colBase + nLo], c[v]);
  }
}

// ---------------- finalize: gather means into padded output ------------------
__global__ void SplitPool_finalize_kernel(const float* __restrict__ sums,
                                          const int* __restrict__ chunk_size,
                                          const int* __restrict__ n_peaks,
                                          const int* __restrict__ pool_start,
                                          float* __restrict__ out,
                                          int max_n_peaks, long out_n) {
  long idx = (long)blockIdx.x * blockDim.x + threadIdx.x;
  if (idx >= out_n) return;
  const int  col = (int)(idx % DIM);
  const long rem = idx / DIM;
  const int  p   = (int)(rem % max_n_peaks);
  const int  b   = (int)(rem / max_n_peaks);

  float val = 0.0f;
  if (p < n_peaks[b]) {
    const int cidx = pool_start[b] + p;
    const int cnt  = max(chunk_size[cidx], 1);
    val = sums[(long)cidx * DIM + col] / (float)cnt;
  }
  out[idx] = val;
}

extern "C" void kernel_launch(void* const* d_in, const int* in_sizes, int n_in,
                              void* d_out, int out_size, void* d_ws, size_t ws_size,
                              hipStream_t stream) {
  const float* x          = (const float*)d_in[0];
  const int*   chunk_size = (const int*)d_in[1];
  const int*   n_peaks    = (const int*)d_in[2];
  (void)n_in; (void)ws_size;

  const int  total_chunks = in_sizes[1];
  const int  batch        = in_sizes[2];
  const long total_rows   = (long)in_sizes[0] / DIM;          // B * L
  const int  max_n_peaks  = out_size / (batch * DIM);

  // workspace layout: [sums: total_chunks*DIM f32][row_start: tc+1][pool_start: batch+1]
  float* sums       = (float*)d_ws;
  int*   row_start  = (int*)(sums + (long)total_chunks * DIM);
  int*   pool_start = row_start + total_chunks + 1;

  SplitPool_prep_kernel<<<1, NTHREADS, 0, stream>>>(
      chunk_size, n_peaks, row_start, pool_start, total_chunks, batch);

  const long zn = (long)total_chunks * DIM;
  SplitPool_zero_kernel<<<(unsigned)((zn + NTHREADS - 1) / NTHREADS), NTHREADS, 0, stream>>>(sums, zn);

  const long ntiles = total_rows / TILE_ROWS;
  SplitPool_segsum_wmma_kernel<<<(unsigned)ntiles, NTHREADS, 0, stream>>>(
      x, row_start, sums, total_chunks);

  const long on = out_size;
  SplitPool_finalize_kernel<<<(unsigned)((on + NTHREADS - 1) / NTHREADS), NTHREADS, 0, stream>>>(
      sums, chunk_size, n_peaks, pool_start, (float*)d_out, max_n_peaks, on);
}